// FusionMamba_32504312496372
// MI455X (gfx1250) — compile-verified
//
#include <hip/hip_runtime.h>
#include <cstdint>

// FusionMamba for MI455X (gfx1250). fp32 throughout; matmuls via
// V_WMMA_F32_16X16X4_F32 (exact fp32 FMA, wave32). Workspace use:
// 16*8192*256 + 2*8192*32 floats ~= 137 MB (L2-resident on MI455X, 192MB L2).

#define ROWS   8192
#define DIM    256
#define DSTATE 8
#define DTRANK 16

typedef __attribute__((ext_vector_type(2))) float v2f;
typedef __attribute__((ext_vector_type(8))) float v8f;

enum { ACT_NONE = 0, ACT_SOFTPLUS = 1, ACT_RELU = 2 };

__device__ __forceinline__ float siluf(float x) { return x / (1.f + __expf(-x)); }

// ---------------------------------------------------------------------------
// LayerNorm: one wave per row (8 elems/lane), optional residual add,
// strided output so we can scatter into a wider (concat) buffer.
// ---------------------------------------------------------------------------
__global__ void ln_kernel(const float* __restrict__ x, const float* __restrict__ g,
                          const float* __restrict__ b, const float* __restrict__ res,
                          float* __restrict__ out, int out_ld, int rows)
{
  const int lane = threadIdx.x & 31;
  const int wave = threadIdx.x >> 5;
  const int row  = blockIdx.x * (blockDim.x >> 5) + wave;
  if (row >= rows) return;
  const float* xr = x + (size_t)row * DIM;
  float v[8];
  float s = 0.f;
#pragma unroll
  for (int i = 0; i < 8; i++) { v[i] = xr[lane + 32 * i]; s += v[i]; }
#pragma unroll
  for (int o = 16; o; o >>= 1) s += __shfl_xor(s, o, 32);
  const float mean = s * (1.f / DIM);
  float var = 0.f;
#pragma unroll
  for (int i = 0; i < 8; i++) { float d = v[i] - mean; var += d * d; }
#pragma unroll
  for (int o = 16; o; o >>= 1) var += __shfl_xor(var, o, 32);
  const float inv = rsqrtf(var * (1.f / DIM) + 1e-5f);
#pragma unroll
  for (int i = 0; i < 8; i++) {
    const int c = lane + 32 * i;
    float o = (v[i] - mean) * inv * g[c] + b[c];
    if (res) o += res[(size_t)row * DIM + c];
    out[(size_t)row * out_ld + c] = o;
  }
}

// ---------------------------------------------------------------------------
// GEMM: Y[rows,M] = act(X[rows,K] * W[M,K]^T + bias), fp32 WMMA 16x16x4.
// Wave tile = 16 rows x 64 cols (4 accumulators). Block = 8 waves = 128 rows.
// A layout (ISA 16x4 f32): lane L holds (X[r0+(L&15)][k+2*(L>>4)], ...+1) -> v2f
// B layout (4x16 f32):     lane L holds (W[m0+(L&15)][k+2*(L>>4)], ...+1) -> v2f
// Out-of-range columns are CLAMPED (not masked): no EXEC divergence in the
// k-loop; clamped lanes feed dead accumulators that are masked at store.
// The k-loop is an explicit 2-stage software pipeline (K % 8 == 0 always):
// loads for step k are issued while WMMAs consume step k-8, so the wait
// before each WMMA group leaves the newest loads in flight.
// ---------------------------------------------------------------------------
__global__ void gemm_wmma_kernel(const float* __restrict__ X, int ldx,
                                 const float* __restrict__ W, int ldw,
                                 const float* __restrict__ bias,
                                 float* __restrict__ Y, int ldy,
                                 int rows, int K, int M, int act)
{
  const int lane = threadIdx.x & 31;
  const int wv   = threadIdx.x >> 5;
  const int half = lane >> 4;        // 0 or 1 (selects K pair)
  const int l    = lane & 15;        // row-of-A / col-of-B within tile
  const int r0   = blockIdx.x * 128 + wv * 16;
  const int m0   = blockIdx.y * 64;

  int arow = r0 + l;
  if (arow >= rows) arow = rows - 1;
  const float* xp = X + (size_t)arow * ldx + 2 * half;

  const float* bp[4];
#pragma unroll
  for (int j = 0; j < 4; j++) {
    int mc = m0 + 16 * j + l;
    if (mc >= M) mc = M - 1;                 // clamp: dead accumulator anyway
    bp[j] = W + (size_t)mc * ldw + 2 * half;
  }

  v8f acc[4];
#pragma unroll
  for (int j = 0; j < 4; j++) acc[j] = (v8f){0.f, 0.f, 0.f, 0.f, 0.f, 0.f, 0.f, 0.f};

  // ---- prologue: stage0 (k=0) and stage1 (k=4) operands ----
  v2f aA = *(const v2f*)(xp + 0);
  v2f bA[4];
#pragma unroll
  for (int j = 0; j < 4; j++) bA[j] = *(const v2f*)(bp[j] + 0);
  v2f aB = *(const v2f*)(xp + 4);
  v2f bB[4];
#pragma unroll
  for (int j = 0; j < 4; j++) bB[j] = *(const v2f*)(bp[j] + 4);

  // ---- steady state: prefetch k / k+4 while consuming k-8 / k-4 ----
  for (int k = 8; k < K; k += 8) {
    v2f aN = *(const v2f*)(xp + k);
    v2f bN[4];
#pragma unroll
    for (int j = 0; j < 4; j++) bN[j] = *(const v2f*)(bp[j] + k);
#pragma unroll
    for (int j = 0; j < 4; j++)
      acc[j] = __builtin_amdgcn_wmma_f32_16x16x4_f32(
          false, aA, false, bA[j], (short)0, acc[j], false, false);
    aA = aN;
#pragma unroll
    for (int j = 0; j < 4; j++) bA[j] = bN[j];

    v2f aM = *(const v2f*)(xp + k + 4);
    v2f bM[4];
#pragma unroll
    for (int j = 0; j < 4; j++) bM[j] = *(const v2f*)(bp[j] + k + 4);
#pragma unroll
    for (int j = 0; j < 4; j++)
      acc[j] = __builtin_amdgcn_wmma_f32_16x16x4_f32(
          false, aB, false, bB[j], (short)0, acc[j], false, false);
    aB = aM;
#pragma unroll
    for (int j = 0; j < 4; j++) bB[j] = bM[j];
  }

  // ---- epilogue: drain the two in-flight stages ----
#pragma unroll
  for (int j = 0; j < 4; j++)
    acc[j] = __builtin_amdgcn_wmma_f32_16x16x4_f32(
        false, aA, false, bA[j], (short)0, acc[j], false, false);
#pragma unroll
  for (int j = 0; j < 4; j++)
    acc[j] = __builtin_amdgcn_wmma_f32_16x16x4_f32(
        false, aB, false, bB[j], (short)0, acc[j], false, false);

  // D layout: VGPR v -> row (r0 + v + 8*half), col (m0 + 16*j + l)
#pragma unroll
  for (int j = 0; j < 4; j++) {
    const int cc = m0 + 16 * j + l;
    if (cc >= M) continue;
    const float bia = bias ? bias[cc] : 0.f;
#pragma unroll
    for (int v = 0; v < 8; v++) {
      const int rr = r0 + v + 8 * half;
      if (rr >= rows) continue;
      float o = acc[j][v] + bia;
      if (act == ACT_SOFTPLUS)      o = (o > 20.f) ? o : log1pf(__expf(o));
      else if (act == ACT_RELU)     o = fmaxf(o, 0.f);
      Y[(size_t)rr * ldy + cc] = o;
    }
  }
}

// ---------------------------------------------------------------------------
// Depthwise causal conv1d (K=4) + SiLU. Reads xm = xz[:, :256] (stride ldx).
// ---------------------------------------------------------------------------
__global__ void conv_silu_kernel(const float* __restrict__ xz, int ldx,
                                 const float* __restrict__ w, const float* __restrict__ b,
                                 float* __restrict__ out, int rows)
{
  const int idx = blockIdx.x * blockDim.x + threadIdx.x;
  if (idx >= rows * DIM) return;
  const int d = idx & (DIM - 1);
  const int t = idx >> 8;
  float acc = b[d];
#pragma unroll
  for (int k = 0; k < 4; k++) {
    const int tt = t + k - 3;
    if (tt >= 0) acc = fmaf(xz[(size_t)tt * ldx + d], w[d * 4 + k], acc);
  }
  out[idx] = siluf(acc);
}

// ---------------------------------------------------------------------------
// Gate: out = scale * (ya [+ yb]) * silu(z), z strided (reads xz[:,256:512]).
// ---------------------------------------------------------------------------
__global__ void gate_kernel(const float* __restrict__ ya, const float* __restrict__ yb,
                            const float* __restrict__ z, int ldz, float scale,
                            float* __restrict__ out, int rows)
{
  const int idx = blockIdx.x * blockDim.x + threadIdx.x;
  if (idx >= rows * DIM) return;
  const int d = idx & (DIM - 1);
  const int t = idx >> 8;
  float v = ya[idx];
  if (yb) v += yb[idx];
  v *= scale;
  out[idx] = v * siluf(z[(size_t)t * ldz + d]);
}

// ---------------------------------------------------------------------------
// Selective scan. lane = s + 8*(d%4); wave handles 4 channels; shfl-xor
// reduce over the 8 states. Up to 4 independent instances batched per launch
// (blockIdx.y) so the latency-bound recurrences overlap. dir=1 => backward.
// Serial chain per step is a single v_fma_f32; exp/mul feed work is hoisted
// by the 4x unroll.
// ---------------------------------------------------------------------------
struct ScanInst {
  const float* x;     // [rows, 256] scan input (conv output)
  const float* dt;    // [rows, 256]
  const float* dbl;   // [rows, 32]: cols 16..23 = B, 24..31 = C
  const float* A_log; // [256, 8]
  const float* D;     // [256]
  float* y;           // [rows, 256] out (includes + x*D)
  int dir;            // 0 fwd, 1 bwd
};

__global__ void scan_kernel(ScanInst i0, ScanInst i1, ScanInst i2, ScanInst i3, int n)
{
  ScanInst I;
  if (blockIdx.y == 0)      I = i0;
  else if (blockIdx.y == 1) I = i1;
  else if (blockIdx.y == 2) I = i2;
  else                      I = i3;

  const int lane = threadIdx.x & 31;
  const int wave = threadIdx.x >> 5;
  const int d = blockIdx.x * 32 + wave * 4 + (lane >> 3);
  const int s = lane & 7;
  const float A  = -__expf(I.A_log[d * DSTATE + s]);
  const float Dd = I.D[d];
  const int dir = I.dir;
  float h = 0.f;
#pragma unroll 4
  for (int i = 0; i < n; i++) {
    const int t = dir ? (n - 1 - i) : i;
    const float dtv = I.dt[(size_t)t * DIM + d];
    const float xv  = I.x [(size_t)t * DIM + d];
    const float Bv  = I.dbl[(size_t)t * 32 + 16 + s];
    const float Cv  = I.dbl[(size_t)t * 32 + 24 + s];
    const float dA  = __expf(dtv * A);
    h = fmaf(dA, h, dtv * Bv * xv);          // the only serial dependence
    float yp = h * Cv;
    yp += __shfl_xor(yp, 1, 32);
    yp += __shfl_xor(yp, 2, 32);
    yp += __shfl_xor(yp, 4, 32);
    if (s == 0) I.y[(size_t)t * DIM + d] = fmaf(xv, Dd, yp);
  }
}

// ---------------------------------------------------------------------------
// Host orchestration
// ---------------------------------------------------------------------------
struct MambaP {
  const float *in_proj, *conv_w, *conv_b, *x_proj, *dt_w, *dt_b,
              *A_log, *D, *out_proj, *ln_g, *ln_b;
};
struct SingleP { const float *pre_g, *pre_b; MambaP m; };
struct CrossP  { const float *n0_g, *n0_b, *n1_g, *n1_b; MambaP m; };

extern "C" void kernel_launch(void* const* d_in, const int* in_sizes, int n_in,
                              void* d_out, int out_size, void* d_ws, size_t ws_size,
                              hipStream_t stream)
{
  (void)in_sizes; (void)n_in; (void)out_size; (void)ws_size;

  // ---- unpack inputs (setup_inputs dict-insertion order, depth-first) ----
  int cur = 0;
  auto nxt = [&]() { return (const float*)d_in[cur++]; };
  const float* pcd_in  = nxt();
  const float* mask_in = nxt();
  auto readMamba = [&](MambaP& mp) {
    mp.in_proj = nxt(); mp.conv_w = nxt(); mp.conv_b = nxt();
    mp.x_proj = nxt();  mp.dt_w = nxt();   mp.dt_b = nxt();
    mp.A_log = nxt();   mp.D = nxt();      mp.out_proj = nxt();
    mp.ln_g = nxt();    mp.ln_b = nxt();
  };
  SingleP pcdL[2], maskL[2];
  for (int i = 0; i < 2; i++) { pcdL[i].pre_g = nxt(); pcdL[i].pre_b = nxt(); readMamba(pcdL[i].m); }
  for (int i = 0; i < 2; i++) { maskL[i].pre_g = nxt(); maskL[i].pre_b = nxt(); readMamba(maskL[i].m); }
  CrossP pcdC, maskC;
  pcdC.n0_g = nxt(); pcdC.n0_b = nxt(); pcdC.n1_g = nxt(); pcdC.n1_b = nxt(); readMamba(pcdC.m);
  maskC.n0_g = nxt(); maskC.n0_b = nxt(); maskC.n1_g = nxt(); maskC.n1_b = nxt(); readMamba(maskC.m);
  const float* fc1_w = nxt(); const float* fc1_b = nxt();
  const float* fc2_w = nxt(); const float* fc2_b = nxt();

  // ---- workspace layout ----
  float* ws = (float*)d_ws;
  const size_t C1 = (size_t)ROWS * DIM;  // 2M floats
  float* Pb  = ws + 0 * C1;
  float* Mb  = ws + 1 * C1;
  float* T0  = ws + 2 * C1;
  float* T1  = ws + 3 * C1;
  float* XZp = ws + 4 * C1;   // 2*C1 (xm | z)
  float* XZm = ws + 6 * C1;   // 2*C1
  float* XCp = ws + 8 * C1;
  float* XCm = ws + 9 * C1;
  float* DTp = ws + 10 * C1;
  float* DTm = ws + 11 * C1;
  float* Y0  = ws + 12 * C1;
  float* Y1  = ws + 13 * C1;
  float* Y2  = ws + 14 * C1;
  float* Y3  = ws + 15 * C1;
  float* DBp = ws + 16 * C1;                       // [ROWS,32]
  float* DBm = DBp + (size_t)ROWS * 32;            // [ROWS,32]
  float* Fbuf = XZp;  // reuse: z of XZp is dead before F is written

  // ---- launch helpers ----
  auto LN = [&](const float* x, const float* g, const float* b, const float* res,
                float* out, int outld) {
    ln_kernel<<<ROWS / 8, 256, 0, stream>>>(x, g, b, res, out, outld, ROWS);
  };
  auto GEMM = [&](const float* X, int ldx, const float* Wm, int ldw, const float* bias,
                  float* Y, int ldy, int K, int M, int act) {
    dim3 grid(ROWS / 128, (M + 63) / 64);
    gemm_wmma_kernel<<<grid, 256, 0, stream>>>(X, ldx, Wm, ldw, bias, Y, ldy, ROWS, K, M, act);
  };
  auto CONV = [&](const float* xz, int ldx, const float* w, const float* b, float* out) {
    conv_silu_kernel<<<(ROWS * DIM) / 256, 256, 0, stream>>>(xz, ldx, w, b, out, ROWS);
  };
  auto GATE = [&](const float* ya, const float* yb, const float* z, int ldz, float sc, float* out) {
    gate_kernel<<<(ROWS * DIM) / 256, 256, 0, stream>>>(ya, yb, z, ldz, sc, out, ROWS);
  };
  auto SCAN = [&](const ScanInst* si, int n_inst) {
    ScanInst a = si[0];
    ScanInst b = (n_inst > 1) ? si[1] : si[0];
    ScanInst c = (n_inst > 2) ? si[2] : si[0];
    ScanInst d = (n_inst > 3) ? si[3] : si[0];
    dim3 grid(DIM / 32, n_inst);
    scan_kernel<<<grid, 256, 0, stream>>>(a, b, c, d, ROWS);
  };

  hipMemcpyAsync(Pb, pcd_in,  C1 * sizeof(float), hipMemcpyDeviceToDevice, stream);
  hipMemcpyAsync(Mb, mask_in, C1 * sizeof(float), hipMemcpyDeviceToDevice, stream);

  // ---- single-block layers (p and m batched through the scan) ----
  for (int li = 0; li < 2; li++) {
    for (int s = 0; s < 2; s++) {
      const SingleP& L = s ? maskL[li] : pcdL[li];
      float* buf = s ? Mb : Pb;
      float* XZ = s ? XZm : XZp;  float* XC = s ? XCm : XCp;
      float* DT = s ? DTm : DTp;  float* DB = s ? DBm : DBp;
      LN(buf, L.pre_g, L.pre_b, nullptr, T0, DIM);
      GEMM(T0, DIM, L.m.in_proj, DIM, nullptr, XZ, 2 * DIM, DIM, 2 * DIM, ACT_NONE);
      CONV(XZ, 2 * DIM, L.m.conv_w, L.m.conv_b, XC);
      GEMM(XC, DIM, L.m.x_proj, DIM, nullptr, DB, 32, DIM, 32, ACT_NONE);
      GEMM(DB, 32, L.m.dt_w, DTRANK, L.m.dt_b, DT, DIM, DTRANK, DIM, ACT_SOFTPLUS);
    }
    ScanInst si[2] = {
      { XCp, DTp, DBp, pcdL[li].m.A_log,  pcdL[li].m.D,  Y0, 0 },
      { XCm, DTm, DBm, maskL[li].m.A_log, maskL[li].m.D, Y1, 0 },
    };
    SCAN(si, 2);
    for (int s = 0; s < 2; s++) {
      const SingleP& L = s ? maskL[li] : pcdL[li];
      float* buf = s ? Mb : Pb;
      float* XZ = s ? XZm : XZp;
      GATE(s ? Y1 : Y0, nullptr, XZ + DIM, 2 * DIM, 1.f, T1);
      GEMM(T1, DIM, L.m.out_proj, DIM, nullptr, T0, DIM, DIM, DIM, ACT_NONE);
      LN(T0, L.m.ln_g, L.m.ln_b, buf, buf, DIM);  // + residual, in place
    }
  }

  // ---- cross blocks (4 scans batched in one launch) ----
  for (int cb = 0; cb < 2; cb++) {
    const CrossP& Cx = cb ? maskC : pcdC;
    float* b0 = cb ? Mb : Pb;  float* b1 = cb ? Pb : Mb;
    float* XZ = cb ? XZm : XZp;  float* XC = cb ? XCm : XCp;
    float* DT = cb ? DTm : DTp;  float* DB = cb ? DBm : DBp;
    LN(b0, Cx.n0_g, Cx.n0_b, nullptr, T0, DIM);
    GEMM(T0, DIM, Cx.m.in_proj, DIM, nullptr, XZ, 2 * DIM, DIM, 2 * DIM, ACT_NONE);
    CONV(XZ, 2 * DIM, Cx.m.conv_w, Cx.m.conv_b, XC);
    LN(b1, Cx.n1_g, Cx.n1_b, nullptr, T0, DIM);
    GEMM(T0, DIM, Cx.m.x_proj, DIM, nullptr, DB, 32, DIM, 32, ACT_NONE);
    GEMM(DB, 32, Cx.m.dt_w, DTRANK, Cx.m.dt_b, DT, DIM, DTRANK, DIM, ACT_SOFTPLUS);
  }
  {
    ScanInst si[4] = {
      { XCp, DTp, DBp, pcdC.m.A_log,  pcdC.m.D,  Y0, 0 },
      { XCp, DTp, DBp, pcdC.m.A_log,  pcdC.m.D,  Y1, 1 },
      { XCm, DTm, DBm, maskC.m.A_log, maskC.m.D, Y2, 0 },
      { XCm, DTm, DBm, maskC.m.A_log, maskC.m.D, Y3, 1 },
    };
    SCAN(si, 4);
  }
  for (int cb = 0; cb < 2; cb++) {
    const CrossP& Cx = cb ? maskC : pcdC;
    float* b0 = cb ? Mb : Pb;
    float* XZ = cb ? XZm : XZp;
    GATE(cb ? Y2 : Y0, cb ? Y3 : Y1, XZ + DIM, 2 * DIM, 0.5f, T1);
    GEMM(T1, DIM, Cx.m.out_proj, DIM, nullptr, T0, DIM, DIM, DIM, ACT_NONE);
    LN(T0, Cx.m.ln_g, Cx.m.ln_b, b0, Fbuf + cb * DIM, 2 * DIM);  // scatter into concat
  }

  // ---- head: relu(F @ fc1^T + b1) @ fc2^T + b2 ----
  GEMM(Fbuf, 2 * DIM, fc1_w, 2 * DIM, fc1_b, T1, DIM, 2 * DIM, DIM, ACT_RELU);
  GEMM(T1, DIM, fc2_w, DIM, fc2_b, (float*)d_out, DIM, DIM, DIM, ACT_NONE);
}